// GaussSplatModel_57191784513784
// MI455X (gfx1250) — compile-verified
//
#include <hip/hip_runtime.h>

#define RES   32
#define GTOT  (RES * RES * RES)   // 32768 gaussians
#define NPTS  256                 // query points
#define NPB   32                  // n-values per block (grid.y = NPTS/NPB = 8)

typedef unsigned int  u32x4 __attribute__((ext_vector_type(4)));
typedef int           i32x8 __attribute__((ext_vector_type(8)));
typedef int           i32x4 __attribute__((ext_vector_type(4)));

__device__ __forceinline__ float fast_rcp(float x) {
#if __has_builtin(__builtin_amdgcn_rcpf)
    return __builtin_amdgcn_rcpf(x);      // v_rcp_f32
#else
    return 1.0f / x;
#endif
}
__device__ __forceinline__ float fast_rsq(float x) {
#if __has_builtin(__builtin_amdgcn_rsqf)
    return __builtin_amdgcn_rsqf(x);      // v_rsq_f32
#else
    return rsqrtf(x);
#endif
}
__device__ __forceinline__ float fast_sigmoid(float v) {
#if __has_builtin(__builtin_amdgcn_tanhf)
    // sigmoid(v) = 0.5*tanh(v/2) + 0.5 : single v_tanh_f32 (gfx1250 trans op)
    return fmaf(0.5f, __builtin_amdgcn_tanhf(0.5f * v), 0.5f);
#else
    return fast_rcp(1.0f + __expf(-v));   // v_exp + v_rcp fallback
#endif
}
__device__ __forceinline__ float softplus_f(float v) {
    // numerically stable log1p(exp(v)) -- setup-only, keep precise
    return fmaxf(v, 0.0f) + log1pf(expf(-fabsf(v)));
}

__global__ void __launch_bounds__(256)
gauss_splat_kernel(const float* __restrict__ points,     // [NPTS,3]
                   const float* __restrict__ positions,  // [G,3]
                   const float* __restrict__ scales,     // [G,3]
                   const float* __restrict__ rotations,  // [G,4]
                   const float* __restrict__ opacity,    // [G,1]
                   const float* __restrict__ sh,         // [G,9]
                   float* __restrict__ out)              // [NPTS*G rgb][G sigma]
{
    __shared__ float  sraw[NPTS * 3];   // TDM landing buffer
    __shared__ float4 spts[NPTS];       // padded (x,y,z,0)/4 for 1x ds_load_b128

    const int tid = threadIdx.x;
    const int g   = blockIdx.x * 256 + tid;

#if defined(__gfx1250__) && __has_builtin(__builtin_amdgcn_tensor_load_to_lds)
    // Stage the points array into LDS via the Tensor Data Mover (wave 0 only).
    if (tid < 32) {
        unsigned long long ga = (unsigned long long)points;                 // 57-bit global addr
        unsigned int       la = (unsigned int)(unsigned long long)&sraw[0]; // LDS byte offset

        // D# group 0: count=1 (valid), lds_addr, global_addr, type=2 ("image")
        u32x4 d0 = { 1u,
                     la,
                     (unsigned int)(ga & 0xFFFFFFFFull),
                     (unsigned int)((ga >> 32) & 0x01FFFFFFull) | (2u << 30) };
        // D# group 1: data_size=4B; tensor_dim0=768; tensor_dim1=1;
        //             tile_dim0=768; tile_dim1=1; tensor_dim0_stride=768
        i32x8 d1 = { (int)(2u << 16),
                     (int)((unsigned)(NPTS * 3) << 16),   // tensor_dim0[15:0] @bits63:48
                     (int)(1u << 16),                     // tensor_dim1[15:0] @bits95:80
                     (int)((unsigned)(NPTS * 3) << 16),   // tile_dim0 @bits127:112
                     1,                                   // tile_dim1=1, tile_dim2=0
                     NPTS * 3,                            // tensor_dim0_stride lo32
                     0, 0 };
        i32x4 d2 = { 0, 0, 0, 0 };
        i32x4 d3 = { 0, 0, 0, 0 };
        i32x8 d4 = { 0, 0, 0, 0, 0, 0, 0, 0 };
        __builtin_amdgcn_tensor_load_to_lds(d0, d1, d2, d3, d4, 0);
        __builtin_amdgcn_s_wait_tensorcnt(0);
    }
    __syncthreads();
    // repack to float4 with points/4 applied (one element per thread)
    spts[tid] = make_float4(sraw[tid * 3 + 0] * 0.25f,
                            sraw[tid * 3 + 1] * 0.25f,
                            sraw[tid * 3 + 2] * 0.25f, 0.0f);
    __syncthreads();
#else
    spts[tid] = make_float4(points[tid * 3 + 0] * 0.25f,
                            points[tid * 3 + 1] * 0.25f,
                            points[tid * 3 + 2] * 0.25f, 0.0f);
    __syncthreads();
#endif

    // ---- per-gaussian setup (reused across the n loop) ----
    const float eps = 1e-6f;
    const float isx = 1.0f / (scales[g * 3 + 0] + eps);
    const float isy = 1.0f / (scales[g * 3 + 1] + eps);
    const float isz = 1.0f / (scales[g * 3 + 2] + eps);
    const float npxs = -positions[g * 3 + 0] * isx;   // so dx = fma(tx, isx, npxs)
    const float npys = -positions[g * 3 + 1] * isy;
    const float npzs = -positions[g * 3 + 2] * isz;

    const float4 q = reinterpret_cast<const float4*>(rotations)[g];
    const float q0 = q.x, q1 = q.y, q2 = q.z, q3 = q.w;
    const float m00 = 1.0f - 2.0f * (q2 * q2 + q3 * q3);
    const float m01 = 2.0f * (q1 * q2 - q0 * q3);
    const float m02 = 2.0f * (q1 * q3 + q0 * q2);
    const float m10 = 2.0f * (q1 * q2 + q0 * q3);
    const float m11 = 1.0f - 2.0f * (q1 * q1 + q3 * q3);
    const float m12 = 2.0f * (q2 * q3 - q0 * q1);
    const float m20 = 2.0f * (q1 * q3 - q0 * q2);
    const float m21 = 2.0f * (q2 * q3 + q0 * q1);
    const float m22 = 1.0f - 2.0f * (q1 * q1 + q2 * q2);

    // SH coefficients pre-scaled by real-SH deg-2 basis constants, with
    // constant terms folded:  a6*(3z^2-1) -> b6*z^2 with -a6 folded into a0f,
    // a8*(x^2-y^2) -> a8*x^2 + c8*y^2.
    const float C0  = 0.28209479177387814f;  // 0.5*sqrt(1/pi)
    const float C1  = 0.48860251190291992f;  // sqrt(3/(4pi))
    const float C2A = 0.54627421529603959f;  // 0.5*sqrt(15/(4pi))
    const float C2B = 0.15769578262626002f;  // 0.5*sqrt(5/(16pi))
    const float C2C = 0.27313710764801980f;  // 0.5*sqrt(15/(16pi))
    const float a1 = C1  * sh[g * 9 + 1];
    const float a2 = C1  * sh[g * 9 + 2];
    const float a3 = C1  * sh[g * 9 + 3];
    const float a4 = C2A * sh[g * 9 + 4];
    const float a5 = C2A * sh[g * 9 + 5];
    const float a6 = C2B * sh[g * 9 + 6];
    const float a7 = C2A * sh[g * 9 + 7];
    const float a8 = C2C * sh[g * 9 + 8];
    const float a0f = C0 * sh[g * 9 + 0] - a6;
    const float b6  = 3.0f * a6;
    const float c8  = -a8;

    // sigma = softplus(opacity), independent of n: one writer slice only
    if (blockIdx.y == 0) {
        out[(size_t)NPTS * GTOT + g] = softplus_f(opacity[g]);
    }

    // ---- main loop over this block's slice of query points ----
    const int n0 = blockIdx.y * NPB;
    unsigned idx = (unsigned)(n0 * GTOT + g);   // fits: N*G = 2^23 < 2^32
    #pragma unroll 4
    for (int n = n0; n < n0 + NPB; ++n) {
        const float4 t = spts[n];               // single ds_load_b128 broadcast

        const float dx = fmaf(t.x, isx, npxs);
        const float dy = fmaf(t.y, isy, npys);
        const float dz = fmaf(t.z, isz, npzs);

        const float rx = fmaf(m00, dx, fmaf(m01, dy, m02 * dz));
        const float ry = fmaf(m10, dx, fmaf(m11, dy, m12 * dz));
        const float rz = fmaf(m20, dx, fmaf(m21, dy, m22 * dz));

        const float inr = fast_rsq(fmaf(rx, rx, fmaf(ry, ry, rz * rz)));
        const float x = rx * inr, y = ry * inr, z = rz * inr;

        float v = a0f;
        v = fmaf(a3, x, v);
        v = fmaf(a1, y, v);
        v = fmaf(a2, z, v);
        v = fmaf(a4, x * y, v);
        v = fmaf(a5, y * z, v);
        v = fmaf(a7, x * z, v);
        v = fmaf(b6, z * z, v);
        v = fmaf(a8, x * x, v);
        v = fmaf(c8, y * y, v);

        out[idx] = fast_sigmoid(v);             // coalesced: consecutive g per lane
        idx += GTOT;
    }
}

extern "C" void kernel_launch(void* const* d_in, const int* in_sizes, int n_in,
                              void* d_out, int out_size, void* d_ws, size_t ws_size,
                              hipStream_t stream) {
    const float* points    = (const float*)d_in[0];
    const float* positions = (const float*)d_in[1];
    const float* scales    = (const float*)d_in[2];
    const float* rotations = (const float*)d_in[3];
    const float* opacity   = (const float*)d_in[4];
    const float* sh        = (const float*)d_in[5];

    dim3 grid(GTOT / 256, NPTS / NPB, 1);   // 128 x 8 workgroups
    gauss_splat_kernel<<<grid, dim3(256, 1, 1), 0, stream>>>(
        points, positions, scales, rotations, opacity, sh, (float*)d_out);
}